// CRFLayer_39900246180199
// MI455X (gfx1250) — compile-verified
//
#include <hip/hip_runtime.h>
#include <hip/hip_bf16.h>
#include <math.h>

// CRF layer for MI455X (gfx1250, wave32).
// B=512, T=2048, K=32.  Forward-algorithm logsumexp recast as sum-product so the
// per-step K x K contraction runs on f32 WMMA (v_wmma_f32_16x16x4_f32).
// Scans are serial in T -> minimize per-step critical path: fast exp/log,
// LDS-resident Viterbi history, tree argmax, software prefetch.
#define B_ 512
#define T_ 2048
#define K_ 32

typedef __attribute__((ext_vector_type(2))) float v2f;
typedef __attribute__((ext_vector_type(8))) float v8f;

__global__ __launch_bounds__(1) void crf_init_loss(float* loss) { loss[0] = 0.0f; }

// ---------------------------------------------------------------------------
// Denominator: forward algorithm.  One wave (32 threads) per 16 batches.
//   m[b]   = max_j S[b,j]
//   P      = exp(S - m)                     (16 x 32, A-matrix fragments)
//   Q      = P @ exp(trans)                 (16 WMMAs: 2 N-tiles x 8 K-chunks)
//   S'[b,j]= m[b] + log Q[b,j] + logits[b,t,j]   (where mask)
// ---------------------------------------------------------------------------
__global__ __launch_bounds__(32) void crf_forward_wmma(
    const float* __restrict__ logits, const unsigned char* __restrict__ mask,
    const float* __restrict__ start_t, const float* __restrict__ end_t,
    const float* __restrict__ trans, float* __restrict__ loss_out)
{
  __shared__ float S[16 * 33];   // row-padded to 33 floats: conflict-free row scans
  __shared__ float Sm[16];

  const int lane = threadIdx.x;        // 0..31
  const int half = lane >> 4;          // 0 | 1  (selects K pair in A/B frags)
  const int lo   = lane & 15;          // M (A) / N (B,C)
  const int b0   = blockIdx.x * 16;

  // B fragments of E = exp(trans): bf0 -> tags 0..15, bf1 -> tags 16..31.
  // B layout mirrors documented A layout: lanes 0-15 hold K=4c,4c+1,
  // lanes 16-31 hold K=4c+2,4c+3; N = lane&15.
  v2f bf0[8], bf1[8];
#pragma unroll
  for (int c = 0; c < 8; ++c) {
    const int k0 = 4 * c + 2 * half;
    bf0[c].x = expf(trans[(k0    ) * K_ + lo]);
    bf0[c].y = expf(trans[(k0 + 1) * K_ + lo]);
    bf1[c].x = expf(trans[(k0    ) * K_ + 16 + lo]);
    bf1[c].y = expf(trans[(k0 + 1) * K_ + 16 + lo]);
  }

  // score0 = start + logits[:,0,:]
  const float st = start_t[lane];
#pragma unroll 4
  for (int b = 0; b < 16; ++b)
    S[b * 33 + lane] = st + logits[((size_t)(b0 + b) * T_) * K_ + lane];
  __syncthreads();

  for (int t = 1; t < T_; ++t) {
    // prefetch next step's emission rows (low occupancy -> SW prefetch)
    if (t + 1 < T_ && lane < 16)
      __builtin_prefetch(&logits[((size_t)(b0 + lane) * T_ + (t + 1)) * K_], 0, 1);

    // per-batch row max (lanes 0..15, one batch each; padded rows -> no conflicts)
    if (lane < 16) {
      float m = S[lane * 33];
#pragma unroll
      for (int i = 1; i < 32; ++i) m = fmaxf(m, S[lane * 33 + i]);
      Sm[lane] = m;
    }
    __syncthreads();

    // A fragments: P[M=lo, k] with k = 4c + 2*half (+1)
    const float mA = Sm[lo];
    v2f a[8];
#pragma unroll
    for (int c = 0; c < 8; ++c) {
      const int k0 = 4 * c + 2 * half;
      a[c].x = __expf(S[lo * 33 + k0    ] - mA);   // args <= 0: fast exp is safe
      a[c].y = __expf(S[lo * 33 + k0 + 1] - mA);
    }

    v8f acc0 = {};
    v8f acc1 = {};
#pragma unroll
    for (int c = 0; c < 8; ++c) {
      acc0 = __builtin_amdgcn_wmma_f32_16x16x4_f32(false, a[c], false, bf0[c],
                                                   (short)0, acc0, false, false);
      acc1 = __builtin_amdgcn_wmma_f32_16x16x4_f32(false, a[c], false, bf1[c],
                                                   (short)0, acc1, false, false);
    }
    __syncthreads();   // all reads of S complete before overwrite

    // C/D layout: VGPR r -> M = half*8 + r, N = lane&15
    const int Mb = half * 8;
#pragma unroll
    for (int r = 0; r < 8; ++r) {
      const int bb = Mb + r;
      const float mrow = Sm[bb];
      const size_t base = ((size_t)(b0 + bb) * T_ + t) * K_;
      if (mask[(size_t)(b0 + bb) * T_ + t]) {
        S[bb * 33 + lo]      = mrow + __logf(acc0[r]) + logits[base + lo];
        S[bb * 33 + 16 + lo] = mrow + __logf(acc1[r]) + logits[base + 16 + lo];
      }
    }
    __syncthreads();
  }

  // denominator[b] = logsumexp_j(S[b,j] + end[j]);  loss += denominator
  if (lane < 16) {
    float m = -3.402823466e38f;
#pragma unroll
    for (int j = 0; j < 32; ++j) m = fmaxf(m, S[lane * 33 + j] + end_t[j]);
    float s = 0.0f;
#pragma unroll
    for (int j = 0; j < 32; ++j) s += __expf(S[lane * 33 + j] + end_t[j] - m);
    atomicAdd(loss_out, m + __logf(s));
  }
}

// ---------------------------------------------------------------------------
// Viterbi: one wave per batch (one block per batch), lane = tag.
// Backpointer history lives entirely in LDS (2047*32 B = 65504 B), so the
// backtrack is a low-latency ds_load_u8 chase and no global history traffic.
// Argmax per step uses a 5-level pairwise tree (short dependency chain).
// ---------------------------------------------------------------------------
__global__ __launch_bounds__(32) void crf_viterbi(
    const float* __restrict__ logits, const float* __restrict__ start_t,
    const float* __restrict__ end_t, const float* __restrict__ trans,
    float* __restrict__ tags_out)
{
  __shared__ unsigned char histS[(T_ - 1) * K_];   // 65504 bytes

  const int lane = threadIdx.x;
  const int b    = blockIdx.x;

  float tr[32];                          // column 'lane' of transitions
#pragma unroll
  for (int i = 0; i < 32; ++i) tr[i] = trans[i * K_ + lane];

  float v = start_t[lane] + logits[(size_t)b * T_ * K_ + lane];

  for (int t = 1; t < T_; ++t) {
    if (t + 4 < T_)
      __builtin_prefetch(&logits[((size_t)b * T_ + (t + 4)) * K_ + lane], 0, 1);

    // all 32 candidates (independent), then pairwise argmax tree.
    float cv[32];
#pragma unroll
    for (int i = 0; i < 32; ++i)
      cv[i] = __shfl(v, i, 32) + tr[i];            // v_readlane broadcast

    float bv[16]; int bi[16];
#pragma unroll
    for (int i = 0; i < 16; ++i) {                 // level 0: 32 -> 16
      const bool g = cv[2 * i + 1] > cv[2 * i];    // '>' keeps smaller index
      bv[i] = g ? cv[2 * i + 1] : cv[2 * i];
      bi[i] = g ? 2 * i + 1 : 2 * i;
    }
#pragma unroll
    for (int w = 8; w >= 1; w >>= 1) {             // levels 1..4
#pragma unroll
      for (int i = 0; i < 8; ++i) {
        if (i < w) {
          const bool g = bv[2 * i + 1] > bv[2 * i];
          bv[i] = g ? bv[2 * i + 1] : bv[2 * i];
          bi[i] = g ? bi[2 * i + 1] : bi[2 * i];
        }
      }
    }
    v = bv[0] + logits[((size_t)b * T_ + t) * K_ + lane];
    histS[(t - 1) * K_ + lane] = (unsigned char)bi[0];
  }

  // argmax_j (v + end[j]) across the wave, smallest index on ties
  float fin = v + end_t[lane];
  int   idx = lane;
#pragma unroll
  for (int off = 16; off > 0; off >>= 1) {
    const float ov = __shfl_xor(fin, off, 32);
    const int   oi = __shfl_xor(idx, off, 32);
    if (ov > fin || (ov == fin && oi < idx)) { fin = ov; idx = oi; }
  }

  __syncthreads();   // fence: histS stores visible before lane-0 chase
  if (lane == 0) {
    int tag = idx;
    tags_out[(size_t)b * T_ + (T_ - 1)] = (float)tag;
    for (int t = T_ - 2; t >= 0; --t) {
      tag = histS[t * K_ + tag];
      tags_out[(size_t)b * T_ + t] = (float)tag;
    }
  }
}

// ---------------------------------------------------------------------------
// Numerator: one block per batch; block-reduce path score + mask count;
// loss -= numerator.
// ---------------------------------------------------------------------------
__global__ __launch_bounds__(256) void crf_numerator(
    const float* __restrict__ logits, const int* __restrict__ labels,
    const unsigned char* __restrict__ mask, const float* __restrict__ start_t,
    const float* __restrict__ end_t, const float* __restrict__ trans,
    float* __restrict__ loss_out)
{
  __shared__ float red[256];
  __shared__ int   redc[256];
  const int b = blockIdx.x;

  float sum = 0.0f;
  int   cnt = 0;
  for (int t = threadIdx.x; t < T_; t += 256) {
    const int   lt = labels[(size_t)b * T_ + t];
    const float e  = logits[((size_t)b * T_ + t) * K_ + lt];
    const int   mk = mask[(size_t)b * T_ + t] ? 1 : 0;
    cnt += mk;
    if (t == 0) {
      sum += start_t[lt] + e;                       // 'first' term (unmasked)
    } else {
      const int lp = labels[(size_t)b * T_ + t - 1];
      sum += (e + trans[lp * K_ + lt]) * (float)mk;
    }
  }
  red[threadIdx.x]  = sum;
  redc[threadIdx.x] = cnt;
  __syncthreads();
  for (int s = 128; s > 0; s >>= 1) {
    if (threadIdx.x < s) {
      red[threadIdx.x]  += red[threadIdx.x + s];
      redc[threadIdx.x] += redc[threadIdx.x + s];
    }
    __syncthreads();
  }
  if (threadIdx.x == 0) {
    const int seq_end = redc[0] - 1;
    const int last    = labels[(size_t)b * T_ + seq_end];
    atomicAdd(loss_out, -(red[0] + end_t[last]));
  }
}

// ---------------------------------------------------------------------------
extern "C" void kernel_launch(void* const* d_in, const int* in_sizes, int n_in,
                              void* d_out, int out_size, void* d_ws, size_t ws_size,
                              hipStream_t stream) {
  (void)in_sizes; (void)n_in; (void)out_size; (void)d_ws; (void)ws_size;
  const float*         logits  = (const float*)d_in[0];
  const int*           labels  = (const int*)d_in[1];
  const unsigned char* mask    = (const unsigned char*)d_in[2];  // jnp bool = 1 byte
  const float*         start_t = (const float*)d_in[3];
  const float*         end_t   = (const float*)d_in[4];
  const float*         trans   = (const float*)d_in[5];

  float* loss = (float*)d_out;       // d_out[0] = loss
  float* tags = loss + 1;            // d_out[1..] = tags (B*T) as f32

  crf_init_loss<<<1, 1, 0, stream>>>(loss);
  crf_forward_wmma<<<B_ / 16, 32, 0, stream>>>(logits, mask, start_t, end_t,
                                               trans, loss);
  crf_viterbi<<<B_, 32, 0, stream>>>(logits, start_t, end_t, trans, tags);
  crf_numerator<<<B_, 256, 0, stream>>>(logits, labels, mask, start_t, end_t,
                                        trans, loss);
}